// QuantumLayer_40699110097347
// MI455X (gfx1250) — compile-verified
//
#include <hip/hip_runtime.h>

// out[b] = (z0, z0*z1, z0*z1*z2, z0*z1*z2*z3),  z_q = cos(x[b,q]) * cos(w[q])
// Derivation: product state before CNOT chain; CNOT conjugation maps
// Z_q -> Z_0..Z_q, and expectations factorize on product states.

typedef float v4f __attribute__((ext_vector_type(4)));

__global__ __launch_bounds__(256) void qlayer_ev_kernel(
    const v4f* __restrict__ x4,      // (B,4) float32, one float4 per sample
    const float* __restrict__ w,     // (4,) float32, uniform -> scalar loads
    v4f* __restrict__ out4,          // (B,4) float32
    int nsamp)
{
    int i = blockIdx.x * 256 + threadIdx.x;
    if (i >= nsamp) return;

    // Shared weights: uniform address, cached in L2/K$; cos is TRANS-unit cheap.
    float cw0 = __builtin_cosf(w[0]);
    float cw1 = __builtin_cosf(w[1]);
    float cw2 = __builtin_cosf(w[2]);
    float cw3 = __builtin_cosf(w[3]);

    // Streaming read: data touched exactly once -> non-temporal hint (TH=NT).
    v4f xv = __builtin_nontemporal_load(&x4[i]);

    float z0 = __builtin_cosf(xv.x) * cw0;
    float z1 = __builtin_cosf(xv.y) * cw1;
    float z2 = __builtin_cosf(xv.z) * cw2;
    float z3 = __builtin_cosf(xv.w) * cw3;

    v4f r;
    r.x = z0;
    r.y = r.x * z1;
    r.z = r.y * z2;
    r.w = r.z * z3;

    // Write-once output -> non-temporal store.
    __builtin_nontemporal_store(r, &out4[i]);
}

extern "C" void kernel_launch(void* const* d_in, const int* in_sizes, int n_in,
                              void* d_out, int out_size, void* d_ws, size_t ws_size,
                              hipStream_t stream) {
    const float* x = (const float*)d_in[0];   // (BATCH, 4) float32
    const float* w = (const float*)d_in[1];   // (4,)      float32
    float* out     = (float*)d_out;           // (BATCH, 4) float32

    int nsamp  = in_sizes[0] / 4;             // BATCH = 524288
    int blocks = (nsamp + 255) / 256;

    qlayer_ev_kernel<<<blocks, 256, 0, stream>>>(
        (const v4f*)x, w, (v4f*)out, nsamp);
}